// Downsample1d_1906965479967
// MI455X (gfx1250) — compile-verified
//
#include <hip/hip_runtime.h>
#include <hip/hip_bf16.h>
#include <stdint.h>

// Downsample1d: depthwise conv1d, K=4 kernel, stride 2, reflect pad 1.
// Memory-bound (AI ~0.7 flop/byte, ~33us HBM floor @23.3TB/s): stream input
// via CDNA5 async global->LDS (ASYNCcnt path), compute 4-tap FMAs from LDS,
// 128-bit non-temporal stores for the output stream.

#define TILE_IN   8192                 // input floats per block tile (32KB LDS)
#define TILE_OUT  (TILE_IN / 2)        // 4096 outputs per tile
#define NTHREADS  256
#define IN_CHUNKS (TILE_IN / (NTHREADS * 4))   // 8 x b128 async loads per lane
#define OUT_VEC4  (TILE_OUT / (4 * NTHREADS))  // 4 x float4 stores per lane

typedef float v4f __attribute__((ext_vector_type(4)));

__global__ __launch_bounds__(NTHREADS)
void downsample1d_kernel(const float* __restrict__ x,
                         const float* __restrict__ kern,
                         float* __restrict__ y,
                         int T) {
    __shared__ __align__(16) float s[TILE_IN];
    __shared__ float sHalo[2];   // [0] = xp[s0], [1] = xp[s0 + TILE_IN + 1]

    const int row  = blockIdx.y;
    const int tile = blockIdx.x;
    const int s0   = tile * TILE_IN;                 // tile start (input index)
    const float* __restrict__ tileBase = x + (long long)row * T + s0;
    const int tid = threadIdx.x;

    // Reflect halos: xp[i] = x[i-1] for 1<=i<=T, xp[0]=x[1], xp[T+1]=x[T-2].
    if (tid == 0) {
        const int li = (s0 == 0)           ? 1             : -1;       // xp[s0]
        const int ri = (s0 + TILE_IN == T) ? (TILE_IN - 2) : TILE_IN;  // xp[s0+TILE+1]
        sHalo[0] = tileBase[li];
        sHalo[1] = tileBase[ri];
    }

    // ---- async global -> LDS bulk copy: s[i] = x[s0 + i], 128b per lane ----
    const uint32_t ldsBase = (uint32_t)(uintptr_t)(&s[0]);
#pragma unroll
    for (int k = 0; k < IN_CHUNKS; ++k) {
        const uint32_t off     = (uint32_t)(k * NTHREADS + tid) * 16u;
        const uint32_t ldsAddr = ldsBase + off;
        // GVS mode: saddr (uniform 64-bit base) + per-lane 32-bit byte offset
        asm volatile("global_load_async_to_lds_b128 %0, %1, %2"
                     :
                     : "v"(ldsAddr), "v"(off), "s"(tileBase)
                     : "memory");
    }
#if __has_builtin(__builtin_amdgcn_s_wait_asynccnt)
    __builtin_amdgcn_s_wait_asynccnt(0);
#else
    asm volatile("s_wait_asynccnt 0x0" ::: "memory");
#endif
    __syncthreads();

    // Weights are uniform -> scalar loads.
    const float w0 = kern[0], w1 = kern[1], w2 = kern[2], w3 = kern[3];

    float* __restrict__ ytile = y + (long long)row * (T / 2) + (s0 / 2);

#pragma unroll
    for (int k = 0; k < OUT_VEC4; ++k) {
        const int v    = k * NTHREADS + tid;  // float4 index in tile's output
        const int base = 8 * v;               // window: lds[base-1 .. base+8]
        float w[10];
        w[0] = (base == 0) ? sHalo[0] : s[base - 1];
#pragma unroll
        for (int i = 1; i <= 8; ++i) w[i] = s[base + i - 1];
        w[9] = (base + 8 >= TILE_IN) ? sHalo[1] : s[base + 8];

        v4f o;
        o.x = w0 * w[0] + w1 * w[1] + w2 * w[2] + w3 * w[3];
        o.y = w0 * w[2] + w1 * w[3] + w2 * w[4] + w3 * w[5];
        o.z = w0 * w[4] + w1 * w[5] + w2 * w[6] + w3 * w[7];
        o.w = w0 * w[6] + w1 * w[7] + w2 * w[8] + w3 * w[9];

        // Streaming output (768MB stream > 192MB L2): non-temporal 128b store.
        __builtin_nontemporal_store(o, (v4f*)(ytile + 4 * v));
    }
}

extern "C" void kernel_launch(void* const* d_in, const int* in_sizes, int n_in,
                              void* d_out, int out_size, void* d_ws, size_t ws_size,
                              hipStream_t stream) {
    const float* x    = (const float*)d_in[0];
    const float* kern = (const float*)d_in[1];
    float*       y    = (float*)d_out;

    const int T    = 131072;                 // time length (reference-fixed)
    const int rows = in_sizes[0] / T;        // B*C = 1024
    if (rows <= 0) return;

    dim3 grid(T / TILE_IN, rows);            // 16 tiles/row x 1024 rows
    downsample1d_kernel<<<grid, NTHREADS, 0, stream>>>(x, kern, y, T);
}